// FrameRigidTransformer_81415400063354
// MI455X (gfx1250) — compile-verified
//
#include <hip/hip_runtime.h>
#include <math.h>

typedef __attribute__((ext_vector_type(2))) float v2f;
typedef __attribute__((ext_vector_type(8))) float v8f;

#define DVOL 96
#define HVOL 256
#define WVOL 256
#define NPTS (256 * 256)

__global__ __launch_bounds__(256)
void frame_rigid_wmma_kernel(const float* __restrict__ vol,
                             const float* __restrict__ dof,
                             float* __restrict__ out) {
    const int lane = threadIdx.x & 31;   // wave32
    const int wave = threadIdx.x >> 5;   // 8 waves per block
    const int tile = blockIdx.x & 255;   // 256 tiles of 256 points per batch
    const int b    = blockIdx.x >> 8;    // batch index

    // ---- per-batch rigid transform matrix (uniform across block) ----
    const float* dd = dof + b * 6;
    const float t0 = dd[0], t1 = dd[1], t2 = dd[2];
    const float DEG2RAD = 0.017453292519943295f;
    const float ai = dd[3] * DEG2RAD, aj = dd[4] * DEG2RAD, ak = dd[5] * DEG2RAD;
    const float si = sinf(ai), sj = sinf(aj), sk = sinf(ak);
    const float ci = cosf(ai), cj = cosf(aj), ck = cosf(ak);
    const float cc = ci * ck, cs = ci * sk, sc = si * ck, ss = si * sk;
    const float m00 = cj * ck, m01 = sj * sc - cs, m02 = sj * cc + ss, m03 = t0;
    const float m10 = cj * sk, m11 = sj * ss + cc, m12 = sj * cs - sc, m13 = t1;
    const float m20 = -sj,     m21 = cj * si,      m22 = cj * ci,      m23 = t2;

    // ---- A operand: 16x4 f32. Matrix rows 0..2 placed at A rows {0,1,2}
    // and duplicated at {8,9,10}. Layout: lane<16 holds A[lane][0],A[lane][1];
    // lane>=16 holds A[lane-16][2],A[lane-16][3]. ----
    const int r  = lane & 15;   // A row this lane contributes
    const int mr = r & 7;       // rows 0,1,2,8,9,10 <=> mr<=2
    v2f A; A.x = 0.0f; A.y = 0.0f;
    if (mr <= 2) {
        if (lane < 16) {
            A.x = (mr == 0) ? m00 : (mr == 1) ? m10 : m20;
            A.y = (mr == 0) ? m01 : (mr == 1) ? m11 : m21;
        } else {
            A.x = (mr == 0) ? m02 : (mr == 1) ? m12 : m22;
            A.y = (mr == 0) ? m03 : (mr == 1) ? m13 : m23;
        }
    }

    // ---- B operands: 4x16 f32, columns = 16 grid points (homogeneous).
    // v0: lanes 0-15 = K0 row (gx), lanes 16-31 = K2 row (gz == 0)
    // v1: lanes 0-15 = K1 row (gy), lanes 16-31 = K3 row (ones)
    // Grid point n: gx = (n%256)-127.5, gy = (n/256)-127.5, gz = 0. ----
    const int base = tile * 256 + wave * 32;   // this wave's 32 points
    v2f B1, B2;
    if (lane < 16) {
        const int p1 = base + lane;
        const int p2 = base + 16 + lane;
        B1.x = (float)(p1 & 255) - 127.5f;
        B1.y = (float)(p1 >> 8)  - 127.5f;
        B2.x = (float)(p2 & 255) - 127.5f;
        B2.y = (float)(p2 >> 8)  - 127.5f;
    } else {
        B1.x = 0.0f; B1.y = 1.0f;
        B2.x = 0.0f; B2.y = 1.0f;
    }

    // ---- coordinate transform on the matrix unit: D = A x B ----
    v8f Z = {};
    v8f D1 = __builtin_amdgcn_wmma_f32_16x16x4_f32(
        false, A, false, B1, (short)0, Z, false, false);
    v8f D2 = __builtin_amdgcn_wmma_f32_16x16x4_f32(
        false, A, false, B2, (short)0, Z, false, false);

    // Lane<16: point base+lane lives in D1 rows 0..2 (VGPR 0..2, N=lane).
    // Lane>=16: point base+lane lives in D2 rows 8..10 (VGPR 0..2, N=lane-16).
    float nx, ny, nz;
    if (lane < 16) { nx = D1[0]; ny = D1[1]; nz = D1[2]; }
    else           { nx = D2[0]; ny = D2[1]; nz = D2[2]; }

    // ---- trilinear sample (normalization collapses to a constant offset) ----
    const float sx = nx + 0.5f * (WVOL - 1);   // +127.5
    const float sy = ny + 0.5f * (HVOL - 1);   // +127.5
    const float sz = nz + 0.5f * (DVOL - 1);   // +47.5

    const float fx0 = floorf(sx), fy0 = floorf(sy), fz0 = floorf(sz);
    const int   x0 = (int)fx0,    y0 = (int)fy0,    z0 = (int)fz0;
    const float fx = sx - fx0,    fy = sy - fy0,    fz = sz - fz0;

    const float* __restrict__ vb = vol + (size_t)b * (DVOL * HVOL * WVOL);
    float acc = 0.0f;
#pragma unroll
    for (int dz = 0; dz < 2; ++dz) {
#pragma unroll
        for (int dy = 0; dy < 2; ++dy) {
#pragma unroll
            for (int dx = 0; dx < 2; ++dx) {
                const int ix = x0 + dx, iy = y0 + dy, iz = z0 + dz;
                const float w = (dx ? fx : 1.0f - fx) *
                                (dy ? fy : 1.0f - fy) *
                                (dz ? fz : 1.0f - fz);
                const bool valid = ((unsigned)ix < (unsigned)WVOL) &&
                                   ((unsigned)iy < (unsigned)HVOL) &&
                                   ((unsigned)iz < (unsigned)DVOL);
                const int cx = min(max(ix, 0), WVOL - 1);
                const int cy = min(max(iy, 0), HVOL - 1);
                const int cz = min(max(iz, 0), DVOL - 1);
                const float g = vb[(size_t)cz * (HVOL * WVOL) + cy * WVOL + cx];
                acc += valid ? g * w : 0.0f;
            }
        }
    }

    out[(size_t)b * NPTS + base + lane] = acc;
}

extern "C" void kernel_launch(void* const* d_in, const int* in_sizes, int n_in,
                              void* d_out, int out_size, void* d_ws, size_t ws_size,
                              hipStream_t stream) {
    (void)n_in; (void)out_size; (void)d_ws; (void)ws_size;
    const float* vol = (const float*)d_in[0];   // (8,1,96,256,256) f32
    const float* dof = (const float*)d_in[1];   // (8,6) f32
    float* out = (float*)d_out;                 // (8,1,1,256,256) f32

    const int B = in_sizes[1] / 6;              // 8 batches
    dim3 grid((unsigned)(B * 256));             // 256 tiles/batch, 256 pts/tile
    dim3 block(256);                            // 8 wave32 waves
    hipLaunchKernelGGL(frame_rigid_wmma_kernel, grid, block, 0, stream,
                       vol, dof, out);
}